// MemConLoss_trans_38079180046961
// MI455X (gfx1250) — compile-verified
//
#include <hip/hip_runtime.h>
#include <hip/hip_bf16.h>
#include <math.h>

typedef __attribute__((ext_vector_type(2))) float v2f;
typedef __attribute__((ext_vector_type(4))) float v4f;
typedef __attribute__((ext_vector_type(8))) float v8f;

#define B_   1024
#define D_   256
#define HW_  49
#define M_   65536
#define INV_T 14.285714285714286f   /* 1/0.07 */
#define COLS_PER_BLOCK 1024
#define NCHUNK (M_ / COLS_PER_BLOCK)  /* 64 */
#define LDA 260                       /* padded LDS stride (floats) */

// ---------------------------------------------------------------------------
// Branchless insert of x into ascending sorted 5-array (drops the largest).
// ---------------------------------------------------------------------------
__device__ __forceinline__ void ins5(float m[5], float x) {
    float t0 = fminf(m[0], x);
    float t1 = fminf(m[1], fmaxf(m[0], x));
    float t2 = fminf(m[2], fmaxf(m[1], x));
    float t3 = fminf(m[3], fmaxf(m[2], x));
    float t4 = fminf(m[4], fmaxf(m[3], x));
    m[0] = t0; m[1] = t1; m[2] = t2; m[3] = t3; m[4] = t4;
}

// ---------------------------------------------------------------------------
// 1) mem_query[b][d] = mean over 7x7 spatial of s_box_feat
// ---------------------------------------------------------------------------
__global__ void k_spatial_mean(const float* __restrict__ box, float* __restrict__ q) {
    int idx = blockIdx.x * blockDim.x + threadIdx.x;
    if (idx >= B_ * D_) return;
    const float* p = box + (size_t)idx * HW_;
    float s = 0.f;
    #pragma unroll
    for (int i = 0; i < HW_; ++i) s += p[i];
    q[idx] = s * (1.0f / (float)HW_);
}

// ---------------------------------------------------------------------------
// 2) row-wise L2 normalize (blockIdx.y selects s_query / mem_s_query)
// ---------------------------------------------------------------------------
__global__ void k_l2norm(const float* __restrict__ sq, const float* __restrict__ msq,
                         float* __restrict__ anc, float* __restrict__ con) {
    __shared__ float red[256];
    int row = blockIdx.x, t = threadIdx.x;
    const float* src = (blockIdx.y == 0) ? sq : msq;
    float* dst       = (blockIdx.y == 0) ? anc : con;
    float v = src[(size_t)row * D_ + t];
    red[t] = v * v;
    __syncthreads();
    for (int s = 128; s > 0; s >>= 1) {
        if (t < s) red[t] += red[t + s];
        __syncthreads();
    }
    float sc = 1.0f / fmaxf(sqrtf(red[0]), 1e-12f);
    dst[(size_t)row * D_ + t] = v * sc;
}

// ---------------------------------------------------------------------------
// 3) GEMM1: logits = (anchor · contrast^T) / T  via V_WMMA_F32_16X16X4_F32
//    one wave per 16x16 tile (only ~1 GFLOP; no need for blocking)
// ---------------------------------------------------------------------------
__global__ __launch_bounds__(32) void k_gemm1(const float* __restrict__ A,
                                              const float* __restrict__ Bm,
                                              float* __restrict__ C) {
    int lane = threadIdx.x;
    int n = lane & 15, h = lane >> 4;
    int rt = blockIdx.y, ct = blockIdx.x;
    const float* ap = A  + (size_t)(rt * 16 + n) * D_ + 2 * h;
    const float* bp = Bm + (size_t)(ct * 16 + n) * D_ + 2 * h;
    v8f acc = {};
    for (int k = 0; k < D_; k += 4) {
        v2f a = *(const v2f*)(ap + k);
        v2f b = *(const v2f*)(bp + k);
        acc = __builtin_amdgcn_wmma_f32_16x16x4_f32(false, a, false, b,
                                                    (short)0, acc, false, false);
    }
    #pragma unroll
    for (int v = 0; v < 8; ++v)
        C[(size_t)(rt * 16 + v + 8 * h) * B_ + ct * 16 + n] = acc[v] * INV_T;
}

// ---------------------------------------------------------------------------
// 4) per-row: max, sum(exp(x-max)), shifted diagonal
// ---------------------------------------------------------------------------
__global__ void k_rowstats(const float* __restrict__ L, float* __restrict__ sumexp,
                           float* __restrict__ smdiag) {
    __shared__ float red[256];
    int row = blockIdx.x, t = threadIdx.x;
    const float* lp = L + (size_t)row * B_;
    float mx = -INFINITY;
    for (int c = t; c < B_; c += 256) mx = fmaxf(mx, lp[c]);
    red[t] = mx; __syncthreads();
    for (int s = 128; s > 0; s >>= 1) {
        if (t < s) red[t] = fmaxf(red[t], red[t + s]);
        __syncthreads();
    }
    float rmax = red[0];
    __syncthreads();
    float se = 0.f;
    for (int c = t; c < B_; c += 256) se += expf(lp[c] - rmax);
    red[t] = se; __syncthreads();
    for (int s = 128; s > 0; s >>= 1) {
        if (t < s) red[t] += red[t + s];
        __syncthreads();
    }
    if (t == 0) { sumexp[row] = red[0]; smdiag[row] = lp[row] - rmax; }
}

// ---------------------------------------------------------------------------
// 5) GEMM2 fused with per-row 5-smallest selection.
//    One wave handles 16 rows x 1024 cols of score = memq · bank^T.
//    Register-blocked: 4 column tiles (64 cols) per K-sweep, so each
//    LDS A-fragment load feeds 4 WMMAs (1.25 loads per WMMA).
//    A tile (16x256 fp32) staged in LDS (padded stride), reused 64x.
// ---------------------------------------------------------------------------
__global__ __launch_bounds__(32) void k_gemm2_min5(const float* __restrict__ Q,
                                                   const float* __restrict__ bank,
                                                   float* __restrict__ part) {
    __shared__ float lA[16 * LDA];
    __shared__ float lM[16][16][5];
    int lane = threadIdx.x;
    int rt = blockIdx.y, cc = blockIdx.x;

    // cooperative load of the 16x256 A tile into LDS (float4, padded stride)
    for (int idx = lane; idx < 16 * 64; idx += 32) {
        int r = idx >> 6, c4 = idx & 63;
        v4f tv = *(const v4f*)(Q + (size_t)(rt * 16 + r) * D_ + c4 * 4);
        *(v4f*)(&lA[r * LDA + c4 * 4]) = tv;
    }
    __syncthreads();

    int n = lane & 15, h = lane >> 4;
    const float* arow = &lA[n * LDA + 2 * h];

    float mins[8][5];
    #pragma unroll
    for (int v = 0; v < 8; ++v)
        #pragma unroll
        for (int q = 0; q < 5; ++q) mins[v][q] = INFINITY;

    // 16 groups of 4 column tiles (64 columns per group)
    for (int g = 0; g < 16; ++g) {
        const float* bp = bank +
            ((size_t)(cc * COLS_PER_BLOCK + g * 64 + n)) * D_ + 2 * h;
        if (g + 1 < 16) __builtin_prefetch(bp + 64 * D_, 0, 1);  // next group
        v8f acc0 = {}, acc1 = {}, acc2 = {}, acc3 = {};
        #pragma unroll 8
        for (int k = 0; k < D_; k += 4) {
            v2f a  = *(const v2f*)(arow + k);
            v2f b0 = *(const v2f*)(bp + k);              //  +0  cols, ioffset 0
            v2f b1 = *(const v2f*)(bp + k + 16 * D_);    //  +16 cols, ioffset 16KB
            v2f b2 = *(const v2f*)(bp + k + 32 * D_);    //  +32 cols, ioffset 32KB
            v2f b3 = *(const v2f*)(bp + k + 48 * D_);    //  +48 cols, ioffset 48KB
            acc0 = __builtin_amdgcn_wmma_f32_16x16x4_f32(false, a, false, b0,
                                                         (short)0, acc0, false, false);
            acc1 = __builtin_amdgcn_wmma_f32_16x16x4_f32(false, a, false, b1,
                                                         (short)0, acc1, false, false);
            acc2 = __builtin_amdgcn_wmma_f32_16x16x4_f32(false, a, false, b2,
                                                         (short)0, acc2, false, false);
            acc3 = __builtin_amdgcn_wmma_f32_16x16x4_f32(false, a, false, b3,
                                                         (short)0, acc3, false, false);
        }
        #pragma unroll
        for (int v = 0; v < 8; ++v) {
            float x0 = acc0[v];               // row = v + 8*h
            float x1 = acc1[v];
            float x2 = acc2[v];
            float x3 = acc3[v];
            if (x0 < mins[v][4]) ins5(mins[v], x0);
            if (x1 < mins[v][4]) ins5(mins[v], x1);
            if (x2 < mins[v][4]) ins5(mins[v], x2);
            if (x3 < mins[v][4]) ins5(mins[v], x3);
        }
    }

    // wave-local merge: 16 lanes x 5 candidates per row -> 5 smallest per row
    #pragma unroll
    for (int v = 0; v < 8; ++v)
        #pragma unroll
        for (int q = 0; q < 5; ++q) lM[8 * h + v][n][q] = mins[v][q];
    __syncthreads();
    if (lane < 16) {
        float mm[5] = {INFINITY, INFINITY, INFINITY, INFINITY, INFINITY};
        for (int s = 0; s < 16; ++s)
            #pragma unroll
            for (int q = 0; q < 5; ++q) {
                float x = lM[lane][s][q];
                if (x < mm[4]) ins5(mm, x);
            }
        size_t base = ((size_t)(rt * 16 + lane) * NCHUNK + cc) * 5;
        #pragma unroll
        for (int q = 0; q < 5; ++q) part[base + q] = mm[q];
    }
}

// ---------------------------------------------------------------------------
// 6) merge per-chunk partials (64x5 per row), compute per-row loss
// ---------------------------------------------------------------------------
__global__ void k_neg_loss(const float* __restrict__ part,
                           const float* __restrict__ sumexp,
                           const float* __restrict__ smdiag,
                           float* __restrict__ loss) {
    int row = blockIdx.x * blockDim.x + threadIdx.x;
    if (row >= B_) return;
    const float* p = part + (size_t)row * NCHUNK * 5;
    float mm[5] = {INFINITY, INFINITY, INFINITY, INFINITY, INFINITY};
    for (int j = 0; j < NCHUNK * 5; ++j) {
        float x = p[j];
        if (x < mm[4]) ins5(mm, x);
    }
    float negsum = expf(mm[0]) + expf(mm[1]) + expf(mm[2]) + expf(mm[3]) + expf(mm[4]);
    loss[row] = -(smdiag[row] - logf(sumexp[row] + negsum));
}

// ---------------------------------------------------------------------------
// 7) mean with exact NaN-guard semantics of the reference
// ---------------------------------------------------------------------------
__global__ void k_final(const float* __restrict__ loss, float* __restrict__ out) {
    __shared__ float rs[256], rz[256];
    int t = threadIdx.x;
    float s = 0.f, z = 0.f;
    for (int i = t; i < B_; i += 256) {
        float v = loss[i];
        s += v;
        z += v * 0.0f;            // nan iff any loss is non-finite (matches ref)
    }
    rs[t] = s; rz[t] = z; __syncthreads();
    for (int k = 128; k > 0; k >>= 1) {
        if (t < k) { rs[t] += rs[t + k]; rz[t] += rz[t + k]; }
        __syncthreads();
    }
    if (t == 0) {
        float m  = rs[0] / (float)B_;
        float fb = rz[0] / (float)B_;
        out[0] = (m != m) ? fb : m;
    }
}

// ---------------------------------------------------------------------------
extern "C" void kernel_launch(void* const* d_in, const int* in_sizes, int n_in,
                              void* d_out, int out_size, void* d_ws, size_t ws_size,
                              hipStream_t stream) {
    const float* s_query  = (const float*)d_in[0];   // [B, D]
    const float* s_box    = (const float*)d_in[1];   // [B, D, 7, 7]
    const float* mem_sq   = (const float*)d_in[2];   // [B, D]
    const float* mem_bank = (const float*)d_in[6];   // [M, D]
    float* out = (float*)d_out;

    float* w      = (float*)d_ws;
    float* memq   = w;                                   // B*D
    float* anc    = memq   + (size_t)B_ * D_;            // B*D
    float* con    = anc    + (size_t)B_ * D_;            // B*D
    float* logits = con    + (size_t)B_ * D_;            // B*B
    float* sumexp = logits + (size_t)B_ * B_;            // B
    float* smdiag = sumexp + B_;                         // B
    float* part   = smdiag + B_;                         // B*NCHUNK*5
    float* loss   = part   + (size_t)B_ * NCHUNK * 5;    // B

    k_spatial_mean<<<dim3((B_ * D_) / 256), 256, 0, stream>>>(s_box, memq);
    k_l2norm<<<dim3(B_, 2), 256, 0, stream>>>(s_query, mem_sq, anc, con);
    k_gemm1<<<dim3(B_ / 16, B_ / 16), 32, 0, stream>>>(anc, con, logits);
    k_rowstats<<<dim3(B_), 256, 0, stream>>>(logits, sumexp, smdiag);
    k_gemm2_min5<<<dim3(NCHUNK, B_ / 16), 32, 0, stream>>>(memq, mem_bank, part);
    k_neg_loss<<<dim3(B_ / 256), 256, 0, stream>>>(part, sumexp, smdiag, loss);
    k_final<<<1, 256, 0, stream>>>(loss, out);
}